// GCN_39754217292624
// MI455X (gfx1250) — compile-verified
//
#include <hip/hip_runtime.h>
#include <math.h>

#define BN_EPS 1e-5f

typedef float v2f __attribute__((ext_vector_type(2)));
typedef float v8f __attribute__((ext_vector_type(8)));

// ---------------------------------------------------------------------------
// small elementwise kernels
// ---------------------------------------------------------------------------
__global__ void k_fill(float* __restrict__ p, float v, int n) {
  int i = blockIdx.x * blockDim.x + threadIdx.x;
  if (i < n) p[i] = v;
}

__global__ void k_count_deg(const int* __restrict__ dst, float* __restrict__ deg, int n_edges) {
  int e = blockIdx.x * blockDim.x + threadIdx.x;
  if (e < n_edges) unsafeAtomicAdd(&deg[dst[e]], 1.0f);
}

__global__ void k_rsqrt_inplace(float* __restrict__ p, int n) {
  int i = blockIdx.x * blockDim.x + threadIdx.x;
  if (i < n) p[i] = rsqrtf(p[i]);  // deg -> D^{-1/2}
}

// ---------------------------------------------------------------------------
// fp32 WMMA GEMM:  Y[N,OUT] = X[N,128] @ W[128,OUT]
// Each wave computes one 16x16 tile via V_WMMA_F32_16X16X4_F32, K-loop of 32.
// block = 256 threads = 8 waves.  OUT=128: 8 col tiles x 1 row tile (16 rows);
// OUT=64: 4 col tiles x 2 row tiles (32 rows).
// ---------------------------------------------------------------------------
template <int OUT>
__global__ __launch_bounds__(256) void k_gemm_wmma(const float* __restrict__ X,
                                                   const float* __restrict__ W,
                                                   float* __restrict__ Y, int n_rows) {
  constexpr int K = 128;
  constexpr int COL_TILES = OUT / 16;        // 8 or 4
  constexpr int ROW_TILES = 8 / COL_TILES;   // 1 or 2
  constexpr int ROWS_PER_BLOCK = ROW_TILES * 16;

  const int wave = threadIdx.x >> 5;
  const int lane = threadIdx.x & 31;
  const int rowTile = wave / COL_TILES;
  const int colTile = wave % COL_TILES;
  const int row0 = blockIdx.x * ROWS_PER_BLOCK + rowTile * 16;
  if (row0 >= n_rows) return;                // uniform within the wave
  const int col0 = colTile * 16;
  const int hi = lane >> 4;                  // 0: K=k+0..1, 1: K=k+2..3
  const int lo = lane & 15;

  int arow = row0 + lo;
  if (arow >= n_rows) arow = n_rows - 1;     // clamp, no divergence
  const float* __restrict__ aptr = X + (size_t)arow * K + 2 * hi;
  const float* __restrict__ bptr = W + (size_t)(2 * hi) * OUT + (col0 + lo);

  v8f acc = {};
#pragma unroll
  for (int kb = 0; kb < K; kb += 4) {
    v2f a = *(const v2f*)(aptr + kb);        // X[row][kb+2hi], X[row][kb+2hi+1]
    v2f b;
    b.x = bptr[(size_t)kb * OUT];            // W[kb+2hi    ][col]
    b.y = bptr[(size_t)kb * OUT + OUT];      // W[kb+2hi + 1][col]
    acc = __builtin_amdgcn_wmma_f32_16x16x4_f32(
        /*neg_a=*/false, a, /*neg_b=*/false, b,
        /*c_mod=*/(short)0, acc, /*reuse_a=*/false, /*reuse_b=*/false);
  }

  // D layout: vgpr i -> row (i + 8*hi), lane lo -> col
  float* __restrict__ yp = Y + (size_t)(row0 + 8 * hi) * OUT + col0 + lo;
#pragma unroll
  for (int i = 0; i < 8; ++i) {
    if (row0 + 8 * hi + i < n_rows) yp[(size_t)i * OUT] = acc[i];
  }
}

// ---------------------------------------------------------------------------
// self-loop + bias init:  agg[i,c] = xw[i,c] * dis[i]^2 + bias[c]
// ---------------------------------------------------------------------------
template <int OUT>
__global__ void k_selfloop_bias(const float* __restrict__ xw, const float* __restrict__ dis,
                                const float* __restrict__ bias, float* __restrict__ agg,
                                int n_nodes) {
  constexpr int SH = (OUT == 128) ? 7 : 6;
  int i = blockIdx.x * blockDim.x + threadIdx.x;
  if (i < (n_nodes << SH)) {
    int node = i >> SH;
    int c = i & (OUT - 1);
    float di = dis[node];
    agg[i] = xw[i] * (di * di) + bias[c];
  }
}

// ---------------------------------------------------------------------------
// edge scatter:  agg[dst] += xw[src] * (dis[src]*dis[dst])
// OUT/4 threads per edge, float4 (b128) gather + 4x global_atomic_add_f32.
// agg buffer (51MB) is L2-resident (192MB L2) so the RMW stays on-chip.
// ---------------------------------------------------------------------------
template <int OUT>
__global__ __launch_bounds__(256) void k_scatter(const int* __restrict__ src,
                                                 const int* __restrict__ dst,
                                                 const float* __restrict__ dis,
                                                 const float* __restrict__ xw,
                                                 float* __restrict__ agg, int n_edges) {
  constexpr int GROUPS = OUT / 4;
  long long tid = (long long)blockIdx.x * blockDim.x + threadIdx.x;
  int e = (int)(tid / GROUPS);
  int g = (int)(tid % GROUPS);
  if (e >= n_edges) return;
  int s = src[e];
  int d = dst[e];
  float norm = dis[s] * dis[d];
  const float4 v = *(const float4*)(xw + (size_t)s * OUT + g * 4);
  float* o = agg + (size_t)d * OUT + g * 4;
  unsafeAtomicAdd(o + 0, v.x * norm);
  unsafeAtomicAdd(o + 1, v.y * norm);
  unsafeAtomicAdd(o + 2, v.z * norm);
  unsafeAtomicAdd(o + 3, v.w * norm);
}

// ---------------------------------------------------------------------------
// BatchNorm: column sums (coalesced: thread c reads column c of each row)
// ---------------------------------------------------------------------------
__global__ void k_bn_stats(const float* __restrict__ x, float* __restrict__ stats,
                           int n_rows, int cols) {
  int c = threadIdx.x;  // blockDim.x == cols
  float s = 0.f, s2 = 0.f;
  for (int r = blockIdx.x; r < n_rows; r += gridDim.x) {
    float v = x[(size_t)r * cols + c];
    s += v;
    s2 += v * v;
  }
  unsafeAtomicAdd(&stats[c], s);
  unsafeAtomicAdd(&stats[cols + c], s2);
}

// fold mean/var/gamma/beta into per-column scale & shift
__global__ void k_bn_coef(float* __restrict__ stats, const float* __restrict__ g,
                          const float* __restrict__ beta, int cols, float invN) {
  int c = threadIdx.x;
  if (c < cols) {
    float m = stats[c] * invN;
    float var = stats[cols + c] * invN - m * m;  // biased batch variance
    float a = g[c] * rsqrtf(var + BN_EPS);
    stats[2 * cols + c] = a;
    stats[3 * cols + c] = beta[c] - m * a;
  }
}

template <int OUT>
__global__ void k_bn_apply_relu(float* __restrict__ x, const float* __restrict__ stats,
                                int total) {
  int i = blockIdx.x * blockDim.x + threadIdx.x;
  if (i < total) {
    int c = i & (OUT - 1);
    float y = stats[2 * OUT + c] * x[i] + stats[3 * OUT + c];
    x[i] = fmaxf(y, 0.0f);
  }
}

// ---------------------------------------------------------------------------
// final projection: out[n] = h[n,:64] . Wf + bf   (one wave32 per node)
// ---------------------------------------------------------------------------
__global__ __launch_bounds__(256) void k_final(const float* __restrict__ h,
                                               const float* __restrict__ Wf,
                                               const float* __restrict__ bf,
                                               float* __restrict__ out, int n) {
  int wave = threadIdx.x >> 5;
  int lane = threadIdx.x & 31;
  int node = blockIdx.x * 8 + wave;
  if (node >= n) return;
  const float* hp = h + (size_t)node * 64;
  float p = hp[lane] * Wf[lane] + hp[lane + 32] * Wf[lane + 32];
#pragma unroll
  for (int off = 16; off > 0; off >>= 1) p += __shfl_down(p, off, 32);
  if (lane == 0) out[node] = p + bf[0];
}

// ---------------------------------------------------------------------------
extern "C" void kernel_launch(void* const* d_in, const int* in_sizes, int n_in,
                              void* d_out, int out_size, void* d_ws, size_t ws_size,
                              hipStream_t stream) {
  const float* x   = (const float*)d_in[0];
  const int*   ei  = (const int*)d_in[1];   // edge_index [2, E]
  const float* W1  = (const float*)d_in[2];
  const float* b1  = (const float*)d_in[3];
  const float* g1  = (const float*)d_in[4];
  const float* be1 = (const float*)d_in[5];
  const float* W2  = (const float*)d_in[6];
  const float* b2  = (const float*)d_in[7];
  const float* g2  = (const float*)d_in[8];
  const float* be2 = (const float*)d_in[9];
  const float* Wf  = (const float*)d_in[10];
  const float* bf  = (const float*)d_in[11];
  float* out = (float*)d_out;

  const int N = in_sizes[0] / 128;
  const int E = in_sizes[1] / 2;
  const int* src = ei;
  const int* dst = ei + E;

  // workspace layout (floats): 2*N*128 + N + 512  (~103 MB)
  float* buf0  = (float*)d_ws;               // xw1 / xw2
  float* buf1  = buf0 + (size_t)N * 128;     // agg1/h1, then agg2/h2
  float* dis   = buf1 + (size_t)N * 128;     // degree -> D^{-1/2}
  float* stats = dis + N;                    // 4 * 128

  const int B = 256;
  // ---- degrees (with self loop) -> rsqrt ----
  k_fill<<<(N + B - 1) / B, B, 0, stream>>>(dis, 1.0f, N);
  k_count_deg<<<(E + B - 1) / B, B, 0, stream>>>(dst, dis, E);
  k_rsqrt_inplace<<<(N + B - 1) / B, B, 0, stream>>>(dis, N);

  // ---- layer 1: GCNConv(128->128) + BN + ReLU ----
  k_gemm_wmma<128><<<(N + 15) / 16, B, 0, stream>>>(x, W1, buf0, N);
  k_selfloop_bias<128><<<(N * 128 + B - 1) / B, B, 0, stream>>>(buf0, dis, b1, buf1, N);
  {
    long long tot = (long long)E * 32;
    k_scatter<128><<<(unsigned)((tot + B - 1) / B), B, 0, stream>>>(src, dst, dis, buf0, buf1, E);
  }
  k_fill<<<2, B, 0, stream>>>(stats, 0.0f, 512);
  k_bn_stats<<<1024, 128, 0, stream>>>(buf1, stats, N, 128);
  k_bn_coef<<<1, 128, 0, stream>>>(stats, g1, be1, 128, 1.0f / (float)N);
  k_bn_apply_relu<128><<<(N * 128 + B - 1) / B, B, 0, stream>>>(buf1, stats, N * 128);

  // ---- layer 2: GCNConv(128->64) + BN + ReLU ----
  k_gemm_wmma<64><<<(N + 31) / 32, B, 0, stream>>>(buf1, W2, buf0, N);
  // buf1 (h1) fully consumed by gemm above; safe to reuse for agg2 (stream order)
  k_selfloop_bias<64><<<(N * 64 + B - 1) / B, B, 0, stream>>>(buf0, dis, b2, buf1, N);
  {
    long long tot = (long long)E * 16;
    k_scatter<64><<<(unsigned)((tot + B - 1) / B), B, 0, stream>>>(src, dst, dis, buf0, buf1, E);
  }
  k_fill<<<2, B, 0, stream>>>(stats, 0.0f, 512);
  k_bn_stats<<<1024, 64, 0, stream>>>(buf1, stats, N, 64);
  k_bn_coef<<<1, 64, 0, stream>>>(stats, g2, be2, 64, 1.0f / (float)N);
  k_bn_apply_relu<64><<<(N * 64 + B - 1) / B, B, 0, stream>>>(buf1, stats, N * 64);

  // ---- final projection ----
  k_final<<<(N + 7) / 8, B, 0, stream>>>(buf1, Wf, bf, out, N);
}